// MTCNN_16045997817915
// MI455X (gfx1250) — compile-verified
//
#include <hip/hip_runtime.h>
#include <hip/hip_bf16.h>
#include <float.h>

typedef float wf2 __attribute__((ext_vector_type(2)));
typedef float wf8 __attribute__((ext_vector_type(8)));

#define SEQ_L 1500
#define WDIM  300
#define LDSZ  450000           // L*D flattened length
#define NFILT 100
#define CHUNK 4096
#define NPT   (CHUNK / 16)     // position tiles per chunk

// ---------------------------------------------------------------------------
// Zero the feats workspace (4 x 300 floats) every launch (graph-replay safe).
// ---------------------------------------------------------------------------
__global__ __launch_bounds__(256) void zero_feats(float* feats) {
  int i = blockIdx.x * 256 + threadIdx.x;
  if (i < 4 * 3 * NFILT) feats[i] = 0.0f;
}

// ---------------------------------------------------------------------------
// Fused embedding-gather -> conv(k=3,4,5) -> bias -> relu -> global max.
// Each block: one batch b, one 4096-wide slice of the flattened sequence.
// Conv expressed as V_WMMA_F32_16X16X4_F32 tiles:
//   D[16 filt x 16 pos] = W[16x4(K taps, zero-padded)] x E[4x16 sliding window]
// k=5 chains a second WMMA (tap 4) using C=D accumulation.
// relu(max_t(y)+bias) == max over relu'd partials -> int-bits atomicMax (>=0).
// ---------------------------------------------------------------------------
__global__ __launch_bounds__(256) void conv_relu_max(
    const int* __restrict__ x, const float* __restrict__ emb,
    const float* __restrict__ w1, const float* __restrict__ b1,
    const float* __restrict__ w2, const float* __restrict__ b2,
    const float* __restrict__ w3, const float* __restrict__ b3,
    float* __restrict__ featsG)   // [4][300], float bits, init 0
{
  __shared__ float se[CHUNK + 4];       // slice of e + halo (k-1 <= 4)
  __shared__ int   sfeats[3 * NFILT];   // block-local relu'd maxima (float bits)

  const int b      = blockIdx.y;
  const int chunk0 = blockIdx.x * CHUNK;
  const int tid    = threadIdx.x;

  for (int i = tid; i < 3 * NFILT; i += 256) sfeats[i] = 0;

  // Stage e[chunk0 .. chunk0+CHUNK+3] into LDS via embedding gather.
  for (int i = tid; i < CHUNK + 4; i += 256) {
    int   t = chunk0 + i;
    float v = 0.0f;
    if (t < LDSZ) {
      int word = t / WDIM;
      int dim  = t - word * WDIM;
      int tok  = x[b * SEQ_L + word];
      v = emb[tok * WDIM + dim];
    }
    se[i] = v;
  }
  __syncthreads();

  const int wave = tid >> 5;
  const int lane = tid & 31;
  const int half = lane >> 4;   // K-split / filter-high half
  const int m    = lane & 15;   // A: filter row; B/D: position column

  // 21 combos = 3 branches (k=3,4,5) x 7 filter tiles (100 -> 7x16 padded)
  for (int combo = 0; combo < 21; ++combo) {
    const int branch = combo / 7;
    const int ftile  = combo % 7;
    const int k      = 3 + branch;
    const float* w    = (branch == 0) ? w1 : (branch == 1) ? w2 : w3;
    const float* bias = (branch == 0) ? b1 : (branch == 1) ? b2 : b3;

    // A operand: A[m][kk] = w[f][kk], kk = 2*half + {0,1}, zero-padded.
    const int fA = ftile * 16 + m;
    wf2 A; A.x = 0.0f; A.y = 0.0f;
    if (fA < NFILT) {
      const int t0 = half * 2;
      A.x = (t0 + 0 < k) ? w[fA * k + t0 + 0] : 0.0f;
      A.y = (t0 + 1 < k) ? w[fA * k + t0 + 1] : 0.0f;
    }
    // Extra A for k=5 tap 4 (K slot 0 only).
    const bool tap5 = (branch == 2);
    wf2 A2; A2.x = 0.0f; A2.y = 0.0f;
    if (tap5 && fA < NFILT && half == 0) A2.x = w[fA * 5 + 4];

    const int tmax = LDSZ - k;   // last valid conv position (inclusive)

    wf8 acc;
    #pragma unroll
    for (int j = 0; j < 8; ++j) acc[j] = -FLT_MAX;

    // Each wave takes position-tiles wave, wave+8, ...
    for (int pt = wave; pt < NPT; pt += 8) {
      const int tloc = pt * 16;
      // B operand: B[kk][n] = e[t0 + n + kk], kk = 2*half + {0,1}
      wf2 Bv;
      Bv.x = se[tloc + m + half * 2 + 0];
      Bv.y = se[tloc + m + half * 2 + 1];
      wf8 c = {};
      wf8 d = __builtin_amdgcn_wmma_f32_16x16x4_f32(
          false, A, false, Bv, (short)0, c, false, false);
      if (tap5) {
        wf2 B2;
        B2.x = (half == 0) ? se[tloc + m + 4] : 0.0f;
        B2.y = 0.0f;
        d = __builtin_amdgcn_wmma_f32_16x16x4_f32(
            false, A2, false, B2, (short)0, d, false, false);
      }
      // This lane owns position column (chunk0 + tloc + m); mask OOB columns.
      const bool valid = (chunk0 + tloc + m) <= tmax;
      #pragma unroll
      for (int j = 0; j < 8; ++j)
        acc[j] = valid ? fmaxf(acc[j], d[j]) : acc[j];
    }

    // Reduce max over the 16 position lanes within each half, then
    // bias + relu and combine into block-local maxima.
    #pragma unroll
    for (int j = 0; j < 8; ++j) {
      float v = acc[j];
      for (int s = 1; s < 16; s <<= 1)
        v = fmaxf(v, __shfl_xor(v, s, 32));
      if (m == 0) {
        const int fm = ftile * 16 + j + 8 * half;  // D filter index
        if (fm < NFILT) {
          float r = fmaxf(v + bias[fm], 0.0f);     // >= 0: int order == float order
          atomicMax(&sfeats[branch * NFILT + fm], __float_as_int(r));
        }
      }
    }
  }

  __syncthreads();
  for (int i = tid; i < 3 * NFILT; i += 256)
    atomicMax((int*)&featsG[b * 300 + i], sfeats[i]);
}

// ---------------------------------------------------------------------------
// Four tiny linear heads: out_i = feats @ fw_i^T + fb_i. 64 outputs total.
// ---------------------------------------------------------------------------
__global__ __launch_bounds__(64) void heads_kernel(
    const float* __restrict__ feats,
    const float* __restrict__ fw1, const float* __restrict__ fb1,
    const float* __restrict__ fw2, const float* __restrict__ fb2,
    const float* __restrict__ fw3, const float* __restrict__ fb3,
    const float* __restrict__ fw4, const float* __restrict__ fb4,
    float* __restrict__ out)
{
  const int tid = threadIdx.x;
  int hs, base;
  const float* fw; const float* fb;
  if (tid < 32)      { hs = 8; base = 0;  fw = fw1; fb = fb1; }
  else if (tid < 40) { hs = 2; base = 32; fw = fw2; fb = fb2; }
  else if (tid < 48) { hs = 2; base = 40; fw = fw3; fb = fb3; }
  else               { hs = 4; base = 48; fw = fw4; fb = fb4; }
  const int rel = tid - base;
  const int b = rel / hs, h = rel % hs;
  float s = fb[h];
  for (int i = 0; i < 3 * NFILT; ++i)
    s += feats[b * 3 * NFILT + i] * fw[h * 3 * NFILT + i];
  out[base + rel] = s;
}

extern "C" void kernel_launch(void* const* d_in, const int* in_sizes, int n_in,
                              void* d_out, int out_size, void* d_ws, size_t ws_size,
                              hipStream_t stream) {
  const int*   x   = (const int*)d_in[0];
  const float* emb = (const float*)d_in[1];
  const float* w1  = (const float*)d_in[2];
  const float* b1  = (const float*)d_in[3];
  const float* w2  = (const float*)d_in[4];
  const float* b2  = (const float*)d_in[5];
  const float* w3  = (const float*)d_in[6];
  const float* b3  = (const float*)d_in[7];
  const float* fw1 = (const float*)d_in[8];
  const float* fb1 = (const float*)d_in[9];
  const float* fw2 = (const float*)d_in[10];
  const float* fb2 = (const float*)d_in[11];
  const float* fw3 = (const float*)d_in[12];
  const float* fb3 = (const float*)d_in[13];
  const float* fw4 = (const float*)d_in[14];
  const float* fb4 = (const float*)d_in[15];
  float* out   = (float*)d_out;
  float* feats = (float*)d_ws;   // 4 x 300 floats = 4.8 KB

  zero_feats<<<dim3(5), 256, 0, stream>>>(feats);
  conv_relu_max<<<dim3((LDSZ + CHUNK - 1) / CHUNK, 4), 256, 0, stream>>>(
      x, emb, w1, b1, w2, b2, w3, b3, feats);
  heads_kernel<<<1, 64, 0, stream>>>(feats, fw1, fb1, fw2, fb2, fw3, fb3,
                                     fw4, fb4, out);
}